// FallbackImpl_46497315947115
// MI455X (gfx1250) — compile-verified
//
#include <hip/hip_runtime.h>

#define MUL  16
#define DD   4
#define S0_  8
#define S1_  8
#define SOUT_ 8
#define P3_  8
#define P4_  4
#define NWAVES 8

typedef float v2f __attribute__((ext_vector_type(2)));
typedef float v4f __attribute__((ext_vector_type(4)));
typedef float v8f __attribute__((ext_vector_type(8)));

// out(e,:) viewed as res[u, c] with c = s_out*4 + k :
//   res(16x32) = A(16x32) @ Wbig(32x32)
//   A[u, s0*4+i]       = x0[i0[e]][s0*64 + u*4 + i]
//   Wbig[s0*4+i, s2*4+k] = sum over paths hitting (s0,s2) of W_p[i,k]
__global__ __launch_bounds__(256)
void tp_wmma_kernel(const float* __restrict__ x0, const int* __restrict__ i0,
                    const float* __restrict__ x1, const float* __restrict__ C3,
                    const float* __restrict__ C4, const int* __restrict__ p3,
                    const int* __restrict__ p4, float* __restrict__ out, int E)
{
    __shared__ alignas(16) float sC3[P3_ * 64];        // C3[p,i,j,k]
    __shared__ alignas(16) float sC4[P4_ * 256];       // C4[p,i,j,k,m]
    __shared__ int   sp3[P3_ * 3];
    __shared__ int   sp4[P4_ * 4];
    __shared__ alignas(16) float swb [NWAVES][32 * 32]; // per-wave Wbig
    __shared__ alignas(16) float sost[NWAVES][512];     // per-wave output staging
    __shared__ alignas(16) float sx1 [NWAVES][32];      // per-wave x1 row

    const int tid  = threadIdx.x;
    const int wave = tid >> 5;
    const int lane = tid & 31;

    // ---- stage small constants once per block ----
    for (int t = tid; t < P3_ * 64;  t += 256) sC3[t] = C3[t];
    for (int t = tid; t < P4_ * 256; t += 256) sC4[t] = C4[t];
    if (tid < P3_ * 3) sp3[tid] = p3[tid];
    if (tid < P4_ * 4) sp4[tid] = p4[tid];
    __syncthreads();

    float* wb  = swb [wave];
    float* ost = sost[wave];
    float* bx  = sx1 [wave];

    const int ml  = lane & 15;   // fragment row/col index (M or N)
    const int khi = lane >> 4;   // K-subslot selector (0/1)
    const int li  = ml >> 2;     // i index 0..3
    const int lk  = ml & 3;      // k (or m) index 0..3

    const int gwaves = gridDim.x * NWAVES;
    for (int e = blockIdx.x * NWAVES + wave; e < E; e += gwaves) {
        // ---- x1 row -> LDS (one float per lane) ----
        bx[lane] = x1[(size_t)e * (S1_ * DD) + lane];

        // ---- zero Wbig (32 lanes x 8 x b128) ----
        v4f z = {0.f, 0.f, 0.f, 0.f};
        #pragma unroll
        for (int t = 0; t < 8; ++t) ((v4f*)wb)[lane + 32 * t] = z;
        asm volatile("s_wait_dscnt 0" ::: "memory");
        __builtin_amdgcn_wave_barrier();

        // ---- accumulate path weight matrices into Wbig ----
        // 3-way paths: two paths per pass, 16 lanes each; lane -> (i,k)
        #pragma unroll
        for (int pb = 0; pb < P3_; pb += 2) {
            int p  = pb + khi;
            int s0 = sp3[p * 3 + 0], s1 = sp3[p * 3 + 1], s2 = sp3[p * 3 + 2];
            const float* c3 = &sC3[p * 64 + li * 16 + lk];   // stride 4 over j
            float w = 0.f;
            #pragma unroll
            for (int j = 0; j < DD; ++j) w += bx[s1 * 4 + j] * c3[j * 4];
            atomicAdd(&wb[(s0 * 4 + li) * 32 + s2 * 4 + lk], w);
        }
        // 4-way paths: lane -> (i,m)
        #pragma unroll
        for (int pb = 0; pb < P4_; pb += 2) {
            int p  = pb + khi;
            int s0 = sp4[p * 4 + 0], s1 = sp4[p * 4 + 1];
            int sc = sp4[p * 4 + 2], s2 = sp4[p * 4 + 3];
            const float* c4 = &sC4[p * 256 + li * 64 + lk];  // [j,k] strides 16,4
            float w = 0.f;
            #pragma unroll
            for (int j = 0; j < DD; ++j) {
                float bj = bx[s1 * 4 + j];
                #pragma unroll
                for (int k = 0; k < DD; ++k)
                    w += bj * bx[sc * 4 + k] * c4[j * 16 + k * 4];
            }
            atomicAdd(&wb[(s0 * 4 + li) * 32 + s2 * 4 + lk], w);
        }
        asm volatile("s_wait_dscnt 0" ::: "memory");
        __builtin_amdgcn_wave_barrier();

        // ---- 16 x V_WMMA_F32_16X16X4_F32 : res = A @ Wbig ----
        const float* arow = x0 + (size_t)i0[e] * 512;
        v8f accLo = {}; v8f accHi = {};
        #pragma unroll
        for (int kc = 0; kc < 8; ++kc) {
            // A fragment (16x4): lane holds A[ml, khi*2 .. khi*2+1] -> b64 load
            v2f a = *(const v2f*)(arow + kc * 64 + ml * 4 + khi * 2);
            // B fragments (4x16) from Wbig rows 4kc..4kc+3
            int r0 = 4 * kc + khi * 2;
            v2f bLo, bHi;
            bLo.x = wb[ r0      * 32 + ml];
            bLo.y = wb[(r0 + 1) * 32 + ml];
            bHi.x = wb[ r0      * 32 + 16 + ml];
            bHi.y = wb[(r0 + 1) * 32 + 16 + ml];
            accLo = __builtin_amdgcn_wmma_f32_16x16x4_f32(
                false, a, false, bLo, (short)0, accLo, false, false);
            accHi = __builtin_amdgcn_wmma_f32_16x16x4_f32(
                false, a, false, bHi, (short)0, accHi, false, false);
        }

        // ---- permute res[u,c] -> out layout (s2, u, k) via LDS staging ----
        #pragma unroll
        for (int g = 0; g < 8; ++g) {
            int u   = g + khi * 8;
            int cLo = ml;
            int cHi = 16 + ml;
            ost[(cLo >> 2) * 64 + u * 4 + (cLo & 3)] = accLo[g];
            ost[(cHi >> 2) * 64 + u * 4 + (cHi & 3)] = accHi[g];
        }
        asm volatile("s_wait_dscnt 0" ::: "memory");
        __builtin_amdgcn_wave_barrier();

        // ---- coalesced b128 stores of the 2KB output row ----
        float* orow = out + (size_t)e * 512;
        #pragma unroll
        for (int q = 0; q < 4; ++q)
            *(v4f*)(orow + q * 128 + lane * 4) =
                *(const v4f*)(ost + q * 128 + lane * 4);
        __builtin_amdgcn_wave_barrier();
    }
}

extern "C" void kernel_launch(void* const* d_in, const int* in_sizes, int n_in,
                              void* d_out, int out_size, void* d_ws, size_t ws_size,
                              hipStream_t stream) {
    const float* x0 = (const float*)d_in[0];
    const int*   i0 = (const int*)  d_in[1];
    const float* x1 = (const float*)d_in[2];
    const float* C3 = (const float*)d_in[3];
    const float* C4 = (const float*)d_in[4];
    const int*   p3 = (const int*)  d_in[5];
    const int*   p4 = (const int*)  d_in[6];
    float* out = (float*)d_out;

    const int E = in_sizes[1];             // i0 has one entry per edge
    int blocks = (E + NWAVES - 1) / NWAVES;
    if (blocks > 4096) blocks = 4096;

    tp_wmma_kernel<<<dim3(blocks), dim3(256), 0, stream>>>(
        x0, i0, x1, C3, C4, p3, p4, out, E);
}